// NeurEPDiff3D_71880572666369
// MI455X (gfx1250) — compile-verified
//
#include <hip/hip_runtime.h>
#include <math.h>

typedef __attribute__((ext_vector_type(2))) float v2f;
typedef __attribute__((ext_vector_type(8))) float v8f;

#define S      135168          // 64*64*33 spatial points per (b,c)
#define BATCH  4
#define WID    20
#define NPTS   (BATCH * S)     // 540672
#define NTILES (NPTS / 16)     // 33792
#define SPEC   2048            // 16*16*8 corner spectrum per (b,c)
#define GRID3  3584            // 16*16*14 real grid per (b,c)
#define NBC    (BATCH * WID)   // 80

#define TWO_PI 6.283185307179586f

__device__ __forceinline__ float gelu_exact(float v) {
    return 0.5f * v * (1.0f + erff(v * 0.70710678118654752f));
}

// ---------------------------------------------------------------------------
// fc0: complex 3 -> 20 pointwise lift (K=3, plain VALU)
// ---------------------------------------------------------------------------
__global__ void k_fc0(const float* __restrict__ xin_r, const float* __restrict__ xin_i,
                      const float* __restrict__ w_r,   const float* __restrict__ w_i,
                      float* __restrict__ xr, float* __restrict__ xi) {
    int q = blockIdx.x * blockDim.x + threadIdx.x;
    if (q >= NPTS) return;
    int b = q / S, p = q - b * S;
    float ar[3], ai[3];
    #pragma unroll
    for (int i = 0; i < 3; ++i) {
        ar[i] = xin_r[(b * 3 + i) * S + p];
        ai[i] = xin_i[(b * 3 + i) * S + p];
    }
    #pragma unroll
    for (int c = 0; c < WID; ++c) {
        float re = 0.f, im = 0.f;
        #pragma unroll
        for (int i = 0; i < 3; ++i) {
            float wr = w_r[i * WID + c], wi = w_i[i * WID + c];
            re += ar[i] * wr - ai[i] * wi;
            im += ar[i] * wi + ai[i] * wr;
        }
        xr[(b * WID + c) * S + p] = re;
        xi[(b * WID + c) * S + p] = im;
    }
}

// ---------------------------------------------------------------------------
// corner gather fused with inverse 16-pt DFT along X
// out[bc, k1, m2, z] = sum_m1 x[bc, gx(m1), gy(m2), z] * e^{+2pi i k1 m1/16}
// ---------------------------------------------------------------------------
__global__ void k_gather_idftx(const float* __restrict__ xr, const float* __restrict__ xi,
                               float* __restrict__ outr, float* __restrict__ outi) {
    int t = blockIdx.x * blockDim.x + threadIdx.x;
    if (t >= NBC * SPEC) return;
    int z  = t & 7;
    int m2 = (t >> 3) & 15;
    int k1 = (t >> 7) & 15;
    int bc = t >> 11;
    int gy = (m2 < 8) ? m2 : m2 + 48;
    const float* br = xr + (size_t)bc * S;
    const float* bi = xi + (size_t)bc * S;
    float accr = 0.f, acci = 0.f;
    for (int m1 = 0; m1 < 16; ++m1) {
        int gx = (m1 < 8) ? m1 : m1 + 48;
        int p  = (gx * 64 + gy) * 33 + z;
        float vr = br[p], vi = bi[p];
        float th = TWO_PI * (float)((k1 * m1) & 15) * (1.0f / 16.0f);
        float s, c; __sincosf(th, &s, &c);
        accr += vr * c - vi * s;
        acci += vr * s + vi * c;
    }
    outr[t] = accr; outi[t] = acci;
}

// ---------------------------------------------------------------------------
// generic 16-pt complex DFT along an axis (stride 8 = Y, stride 128 = X)
// sign = +1 inverse, -1 forward
// ---------------------------------------------------------------------------
__global__ void k_dft16(const float* __restrict__ inr, const float* __restrict__ ini,
                        float* __restrict__ outr, float* __restrict__ outi,
                        int stride, float sign) {
    int t = blockIdx.x * blockDim.x + threadIdx.x;
    if (t >= NBC * SPEC) return;
    int k = (t / stride) & 15;
    int base = t - k * stride;
    float accr = 0.f, acci = 0.f;
    for (int n = 0; n < 16; ++n) {
        float vr = inr[base + n * stride], vi = ini[base + n * stride];
        float th = sign * TWO_PI * (float)((k * n) & 15) * (1.0f / 16.0f);
        float s, c; __sincosf(th, &s, &c);
        accr += vr * c - vi * s;
        acci += vr * s + vi * c;
    }
    outr[t] = accr; outi[t] = acci;
}

// ---------------------------------------------------------------------------
// irfft along Z: 8 complex -> 14 real, fold total 1/(16*16*14) scaling here
// ---------------------------------------------------------------------------
__global__ void k_irfftz(const float* __restrict__ inr, const float* __restrict__ ini,
                         float* __restrict__ g) {
    int t = blockIdx.x * blockDim.x + threadIdx.x;
    if (t >= NBC * GRID3) return;
    int n = t % 14;
    int rest = t / 14;
    const float* cr = inr + rest * 8;
    const float* ci = ini + rest * 8;
    float acc = 0.f;
    for (int k = 0; k < 8; ++k) {
        float w = (k == 0 || k == 7) ? 1.0f : 2.0f;
        float th = TWO_PI * (float)((k * n) % 14) * (1.0f / 14.0f);
        float s, c; __sincosf(th, &s, &c);
        acc += w * (cr[k] * c - ci[k] * s);
    }
    g[t] = acc * (1.0f / 3584.0f);
}

// ---------------------------------------------------------------------------
// per-point 20x20 real channel mix (weights differ per spatial point)
// ---------------------------------------------------------------------------
__global__ void k_einsum(const float* __restrict__ g, const float* __restrict__ w,
                         float* __restrict__ g2) {
    int t = blockIdx.x * blockDim.x + threadIdx.x;
    if (t >= BATCH * WID * GRID3) return;
    int p3 = t % GRID3;
    int o  = (t / GRID3) % WID;
    int b  = t / (GRID3 * WID);
    float acc = 0.f;
    for (int i = 0; i < WID; ++i)
        acc += g[(b * WID + i) * GRID3 + p3] * w[(i * WID + o) * GRID3 + p3];
    g2[t] = acc;
}

// ---------------------------------------------------------------------------
// rfft along Z: 14 real -> 8 complex
// ---------------------------------------------------------------------------
__global__ void k_rfftz(const float* __restrict__ g2,
                        float* __restrict__ outr, float* __restrict__ outi) {
    int t = blockIdx.x * blockDim.x + threadIdx.x;
    if (t >= NBC * SPEC) return;
    int k = t & 7;
    int rest = t >> 3;
    const float* yv = g2 + rest * 14;
    float accr = 0.f, acci = 0.f;
    for (int n = 0; n < 14; ++n) {
        float th = TWO_PI * (float)((k * n) % 14) * (1.0f / 14.0f);
        float s, c; __sincosf(th, &s, &c);
        accr += yv[n] * c;
        acci -= yv[n] * s;
    }
    outr[t] = accr; outi[t] = acci;
}

// ---------------------------------------------------------------------------
// Layer mix (WMMA fp32): x = act((wpt . x + scatter(ym)) * smooth), in place.
// One wave per 16-point tile, both 16-channel N-tiles; complex GEMM as real GEMM, K=40:
//   yr = [xr|xi] . [Wr; -Wi],  yi = [xr|xi] . [Wi; Wr]
// A fragments (20 floats/lane = all inputs of the tile) loaded once, reused
// for both components and both N-tiles. B hoisted into regs across the loop.
// ---------------------------------------------------------------------------
__global__ void __launch_bounds__(256) k_mix_wmma(
    float* __restrict__ xr, float* __restrict__ xi,
    const float* __restrict__ wpt_r, const float* __restrict__ wpt_i,
    const float* __restrict__ ymr, const float* __restrict__ ymi,
    const float* __restrict__ smooth, int layer, int do_gelu)
{
    int lane = threadIdx.x & 31;
    int wave = (blockIdx.x * blockDim.x + threadIdx.x) >> 5;
    int nwaves = (gridDim.x * blockDim.x) >> 5;
    int nloc = lane & 15;
    int half = lane >> 4;

    const float* Wr = wpt_r + layer * WID * WID;
    const float* Wi = wpt_i + layer * WID * WID;

    // B fragments: [comp: 0=re-out 1=im-out][ntile][K-chunk]
    v2f Bf[2][2][10];
    #pragma unroll
    for (int nt = 0; nt < 2; ++nt) {
        int o = nt * 16 + nloc;
        bool ov = (o < WID);
        #pragma unroll
        for (int kc = 0; kc < 10; ++kc) {
            #pragma unroll
            for (int e = 0; e < 2; ++e) {
                int k = kc * 4 + half * 2 + e;
                float vr = 0.f, vi = 0.f;
                if (ov) {
                    if (k < WID) { vr =  Wr[k * WID + o]; vi = Wi[k * WID + o]; }
                    else { int i = k - WID; vr = -Wi[i * WID + o]; vi = Wr[i * WID + o]; }
                }
                Bf[0][nt][kc][e] = vr;
                Bf[1][nt][kc][e] = vi;
            }
        }
    }

    for (int tile = wave; tile < NTILES; tile += nwaves) {
        int q0 = tile * 16;
        int b = q0 / S;
        int pbase = q0 - b * S;
        const float* rowr = xr + (size_t)b * WID * S;
        const float* rowi = xi + (size_t)b * WID * S;
        int pm = pbase + nloc;

        v2f Af[10];
        #pragma unroll
        for (int kc = 0; kc < 10; ++kc) {
            int k0 = kc * 4 + half * 2;
            const float* src = (k0 < WID) ? rowr : rowi;
            int c0 = (k0 < WID) ? k0 : k0 - WID;
            Af[kc][0] = src[c0 * S + pm];
            Af[kc][1] = src[(c0 + 1) * S + pm];
        }

        #pragma unroll
        for (int nt = 0; nt < 2; ++nt) {
            v8f accR = {}; v8f accI = {};
            #pragma unroll
            for (int kc = 0; kc < 10; ++kc) {
                accR = __builtin_amdgcn_wmma_f32_16x16x4_f32(
                    false, Af[kc], false, Bf[0][nt][kc], (short)0, accR, false, false);
                accI = __builtin_amdgcn_wmma_f32_16x16x4_f32(
                    false, Af[kc], false, Bf[1][nt][kc], (short)0, accI, false, false);
            }
            int o = nt * 16 + nloc;
            if (o < WID) {
                #pragma unroll
                for (int vg = 0; vg < 8; ++vg) {
                    int m = vg + half * 8;
                    int p = pbase + m;
                    float re = accR[vg], im = accI[vg];
                    int z  = p % 33;
                    int y  = (p / 33) & 63;
                    int xx = p / (33 * 64);
                    if (z < 8 && (y < 8 || y >= 56) && (xx < 8 || xx >= 56)) {
                        int m1 = (xx < 8) ? xx : xx - 48;
                        int m2 = (y  < 8) ? y  : y  - 48;
                        int si = (b * WID + o) * SPEC + (m1 * 16 + m2) * 8 + z;
                        re += ymr[si]; im += ymi[si];
                    }
                    float sm = smooth[p];
                    re *= sm; im *= sm;
                    if (do_gelu) { re = gelu_exact(re); im = gelu_exact(im); }
                    xr[((size_t)b * WID + o) * S + p] = re;
                    xi[((size_t)b * WID + o) * S + p] = im;
                }
            }
        }
    }
}

// ---------------------------------------------------------------------------
// Fused head: fc1 (20->128, WMMA) -> cgelu -> fc2 (128->3, VALU from LDS).
// 128-wide complex intermediate stays in a 16KB/wave LDS slab (saves ~1.1 GB
// of HBM traffic). 4 waves/block -> 64 KB static LDS.
// ---------------------------------------------------------------------------
#define FC_WAVES 4
__global__ void __launch_bounds__(128) k_fc_wmma(
    const float* __restrict__ xr, const float* __restrict__ xi,
    const float* __restrict__ fc1_r, const float* __restrict__ fc1_i,
    const float* __restrict__ fc2_r, const float* __restrict__ fc2_i,
    float* __restrict__ out)
{
    __shared__ float acts[FC_WAVES][16][128][2];
    int lane = threadIdx.x & 31;
    int wv   = threadIdx.x >> 5;
    int nloc = lane & 15;
    int half = lane >> 4;
    int tileStride = gridDim.x * FC_WAVES;

    for (int tile = blockIdx.x * FC_WAVES + wv; tile < NTILES; tile += tileStride) {
        int q0 = tile * 16;
        int b = q0 / S;
        int pbase = q0 - b * S;
        const float* rowr = xr + (size_t)b * WID * S;
        const float* rowi = xi + (size_t)b * WID * S;
        int pm = pbase + nloc;

        v2f Af[10];
        #pragma unroll
        for (int kc = 0; kc < 10; ++kc) {
            int k0 = kc * 4 + half * 2;
            const float* src = (k0 < WID) ? rowr : rowi;
            int c0 = (k0 < WID) ? k0 : k0 - WID;
            Af[kc][0] = src[c0 * S + pm];
            Af[kc][1] = src[(c0 + 1) * S + pm];
        }

        for (int ht = 0; ht < 8; ++ht) {
            int h = ht * 16 + nloc;
            v2f BR[10], BI[10];
            #pragma unroll
            for (int kc = 0; kc < 10; ++kc) {
                #pragma unroll
                for (int e = 0; e < 2; ++e) {
                    int k = kc * 4 + half * 2 + e;
                    float vr, vi;
                    if (k < WID) { vr =  fc1_r[k * 128 + h]; vi = fc1_i[k * 128 + h]; }
                    else { int i = k - WID; vr = -fc1_i[i * 128 + h]; vi = fc1_r[i * 128 + h]; }
                    BR[kc][e] = vr; BI[kc][e] = vi;
                }
            }
            v8f aR = {}; v8f aI = {};
            #pragma unroll
            for (int kc = 0; kc < 10; ++kc) {
                aR = __builtin_amdgcn_wmma_f32_16x16x4_f32(
                    false, Af[kc], false, BR[kc], (short)0, aR, false, false);
                aI = __builtin_amdgcn_wmma_f32_16x16x4_f32(
                    false, Af[kc], false, BI[kc], (short)0, aI, false, false);
            }
            #pragma unroll
            for (int vg = 0; vg < 8; ++vg) {
                int m = vg + half * 8;
                acts[wv][m][h][0] = gelu_exact(aR[vg]);
                acts[wv][m][h][1] = gelu_exact(aI[vg]);
            }
        }
        __syncthreads();

        for (int idx = lane; idx < 48; idx += 32) {
            int m = idx / 3;
            int j = idx - m * 3;
            float re = 0.f, im = 0.f;
            for (int h = 0; h < 128; ++h) {
                float ar = acts[wv][m][h][0], ai = acts[wv][m][h][1];
                float wr = fc2_r[h * 3 + j], wi = fc2_i[h * 3 + j];
                re += ar * wr - ai * wi;
                im += ar * wi + ai * wr;
            }
            int p = pbase + m;
            size_t oi = ((size_t)(b * 3 + j) * S + p) * 2;
            out[oi]     = re;
            out[oi + 1] = im;
        }
        __syncthreads();
    }
}

// ---------------------------------------------------------------------------
// host launch
// ---------------------------------------------------------------------------
extern "C" void kernel_launch(void* const* d_in, const int* in_sizes, int n_in,
                              void* d_out, int out_size, void* d_ws, size_t ws_size,
                              hipStream_t stream) {
    (void)in_sizes; (void)n_in; (void)out_size; (void)ws_size;

    const float* x_r    = (const float*)d_in[0];
    const float* x_i    = (const float*)d_in[1];
    const float* smooth = (const float*)d_in[2];
    const float* wconv  = (const float*)d_in[3];
    const float* wpt_r  = (const float*)d_in[4];
    const float* wpt_i  = (const float*)d_in[5];
    const float* fc0_r  = (const float*)d_in[6];
    const float* fc0_i  = (const float*)d_in[7];
    const float* fc1_r  = (const float*)d_in[8];
    const float* fc1_i  = (const float*)d_in[9];
    const float* fc2_r  = (const float*)d_in[10];
    const float* fc2_i  = (const float*)d_in[11];
    float* out = (float*)d_out;

    // workspace partition (floats)
    float* w = (float*)d_ws;
    const size_t act = (size_t)NBC * S;      // 10,813,440
    float* xr  = w;            w += act;
    float* xi  = w;            w += act;
    float* c0r = w;            w += NBC * SPEC;
    float* c0i = w;            w += NBC * SPEC;
    float* c1r = w;            w += NBC * SPEC;
    float* c1i = w;            w += NBC * SPEC;
    float* g   = w;            w += NBC * GRID3;
    float* g2  = w;            w += NBC * GRID3;
    float* ymr = w;            w += NBC * SPEC;
    float* ymi = w;            w += NBC * SPEC;

    // lift 3 -> 20
    k_fc0<<<NPTS / 256, 256, 0, stream>>>(x_r, x_i, fc0_r, fc0_i, xr, xi);

    const int gSpec = (NBC * SPEC) / 256;    // 640
    const int gGrid = (NBC * GRID3) / 256;   // 1120

    for (int l = 0; l < 4; ++l) {
        // spectral conv on the 4 low-frequency corner blocks
        k_gather_idftx<<<gSpec, 256, 0, stream>>>(xr, xi, c0r, c0i);
        k_dft16<<<gSpec, 256, 0, stream>>>(c0r, c0i, c1r, c1i, 8,  +1.0f);   // inv Y
        k_irfftz<<<gGrid, 256, 0, stream>>>(c1r, c1i, g);                    // inv Z
        k_einsum<<<gGrid, 256, 0, stream>>>(g, wconv + (size_t)l * WID * WID * GRID3, g2);
        k_rfftz<<<gSpec, 256, 0, stream>>>(g2, c0r, c0i);                    // fwd Z
        k_dft16<<<gSpec, 256, 0, stream>>>(c0r, c0i, c1r, c1i, 8,  -1.0f);   // fwd Y
        k_dft16<<<gSpec, 256, 0, stream>>>(c1r, c1i, ymr, ymi, 128, -1.0f);  // fwd X

        // in-place pointwise mix + corner add + smooth (+ cgelu for l<3)
        k_mix_wmma<<<528, 256, 0, stream>>>(xr, xi, wpt_r, wpt_i, ymr, ymi,
                                            smooth, l, (l < 3) ? 1 : 0);
    }

    // fused fc1 -> cgelu -> fc2 head, complex interleaved output
    k_fc_wmma<<<264, 128, 0, stream>>>(xr, xi, fc1_r, fc1_i, fc2_r, fc2_i, out);
}